// SpatialTransformer_52424370815166
// MI455X (gfx1250) — compile-verified
//
#include <hip/hip_runtime.h>

// 3D trilinear spatial-transformer warp, MI455X (gfx1250).
// Memory-bound gather/stream kernel (AI ~2 flop/byte -> HBM-limited at 23.3TB/s).
// - NT-hint the 165MB flow stream + 55MB output so the 55MB volume stays
//   resident in the 192MB L2 (load/store TH=NT per ISA §7.3).
// - b128 vector loads/stores for the flow/output streams.
// - global_prefetch_b8 read-ahead on the flow planes.
// - All gather addresses are uniform-SGPR-base + unsigned 32-bit VGPR offsets
//   so the backend uses GVS (saddr) addressing: no 64-bit VGPR address adds.
// - Launch geometry carries the (b,d,h,w) decomposition: zero divisions.

typedef float v4f __attribute__((ext_vector_type(4)));

#define DV 160
#define HV 192
#define WV 224
#define PLANE (DV * HV * WV)   // 6,881,280 elements per channel-plane
#define W4 (WV / 4)            // 56 float4-quads per row
#define HSUB 4                 // h-rows per block

__global__ __launch_bounds__(W4 * HSUB) void SpatialTransformer_52424370815166_kernel(
    const float* __restrict__ vol,   // [B,1,D,H,W]
    const float* __restrict__ trf,   // [B,3,D,H,W]
    float* __restrict__ out)         // [B,1,D,H,W]
{
    // block (56,4): tx = quad-in-row, ty = sub-row. 224 threads = 7 wave32.
    const int w0 = threadIdx.x * 4;
    const int h  = blockIdx.x * HSUB + threadIdx.y;
    const int d  = blockIdx.y;
    const int b  = blockIdx.z;

    const unsigned row = (unsigned)((d * HV + h) * WV + w0); // 16B-aligned quad
    const float* __restrict__ volb = vol + (long)b * PLANE;  // uniform bases
    const float* __restrict__ trfb = trf + (long)b * 3L * PLANE;
    float* __restrict__ outb       = out + (long)b * PLANE;

    // Streaming (non-temporal) loads of the three displacement planes:
    // read-once data must not evict the L2-resident volume.
    v4f fd = __builtin_nontemporal_load((const v4f*)(trfb + row));
    v4f fh = __builtin_nontemporal_load((const v4f*)(trfb + (PLANE + row)));
    v4f fw = __builtin_nontemporal_load((const v4f*)(trfb + (2u * PLANE + row)));

    // gfx1250 read-ahead on the flow streams (emits global_prefetch_b8).
    __builtin_prefetch(trfb + (row + 4096u), 0, 1);
    __builtin_prefetch(trfb + (PLANE + row + 4096u), 0, 1);
    __builtin_prefetch(trfb + (2u * PLANE + row + 4096u), 0, 1);

    v4f res;
#pragma unroll
    for (int j = 0; j < 4; ++j) {
        // loc = clip(grid + flow, 0, max)  (matches reference exactly)
        float ld = fminf(fmaxf((float)d        + fd[j], 0.f), (float)(DV - 1));
        float lh = fminf(fmaxf((float)h        + fh[j], 0.f), (float)(HV - 1));
        float lw = fminf(fmaxf((float)(w0 + j) + fw[j], 0.f), (float)(WV - 1));

        // frac weights: identical result to reference's float(i1)-loc form
        // (they differ only at loc==max, where both corners alias anyway).
        float fd0 = floorf(ld), fh0 = floorf(lh), fx0 = floorf(lw);
        float td = ld - fd0, sd = 1.f - td;   // td -> ceil, sd -> floor
        float th = lh - fh0, sh = 1.f - th;
        float tw = lw - fx0, sw = 1.f - tw;

        int d0 = (int)fd0, h0 = (int)fh0, x0 = (int)fx0;
        int d1 = min(d0 + 1, DV - 1);
        int h1 = min(h0 + 1, HV - 1);
        int x1 = min(x0 + 1, WV - 1);

        // 32-bit element offsets from the uniform volb base (GVS addressing)
        unsigned r00 = (unsigned)((d0 * HV + h0) * WV);
        unsigned r01 = (unsigned)((d0 * HV + h1) * WV);
        unsigned r10 = (unsigned)((d1 * HV + h0) * WV);
        unsigned r11 = (unsigned)((d1 * HV + h1) * WV);
        unsigned ux0 = (unsigned)x0, ux1 = (unsigned)x1;

        float v000 = volb[r00 + ux0], v001 = volb[r00 + ux1];
        float v010 = volb[r01 + ux0], v011 = volb[r01 + ux1];
        float v100 = volb[r10 + ux0], v101 = volb[r10 + ux1];
        float v110 = volb[r11 + ux0], v111 = volb[r11 + ux1];

        res[j] = sd * (sh * (sw * v000 + tw * v001) +
                       th * (sw * v010 + tw * v011)) +
                 td * (sh * (sw * v100 + tw * v101) +
                       th * (sw * v110 + tw * v111));
    }

    // Non-temporal store: output is write-once, keep it out of the way of vol.
    __builtin_nontemporal_store(res, (v4f*)(outb + row));
}

extern "C" void kernel_launch(void* const* d_in, const int* in_sizes, int n_in,
                              void* d_out, int out_size, void* d_ws, size_t ws_size,
                              hipStream_t stream) {
    (void)in_sizes; (void)n_in; (void)d_ws; (void)ws_size; (void)out_size;
    const float* vol = (const float*)d_in[0];   // [2,1,160,192,224] f32
    const float* trf = (const float*)d_in[1];   // [2,3,160,192,224] f32
    float* out       = (float*)d_out;           // [2,1,160,192,224] f32

    dim3 block(W4, HSUB, 1);                    // 224 threads = 7 wave32
    dim3 grid(HV / HSUB, DV, 2);                // (48, 160, 2) — exact fit
    SpatialTransformer_52424370815166_kernel<<<grid, block, 0, stream>>>(vol, trf, out);
}